// LSTM_69114613728655
// MI455X (gfx1250) — compile-verified
//
#include <hip/hip_runtime.h>
#include <hip/hip_bf16.h>
#include <stdint.h>

// ---------------- problem dims ----------------
#define B_   512
#define T_   1024
#define I_   50
#define H_   32
#define G4H  128           // 4*H
#define KPAD 52            // I padded to multiple of 4

typedef __attribute__((ext_vector_type(2))) float v2f;
typedef __attribute__((ext_vector_type(8))) float v8f;

// D = A(16x4) * B(4x16) + C(16x16), fp32, wave32
__device__ __forceinline__ v8f wmma_f32_k4(v2f a, v2f b, v8f c) {
  return __builtin_amdgcn_wmma_f32_16x16x4_f32(
      /*neg_a=*/false, a, /*neg_b=*/false, b,
      /*c_mod=*/(short)0, c, /*reuse_a=*/false, /*reuse_b=*/false);
}

__device__ __forceinline__ float sigm(float x) {
  return 1.0f / (1.0f + __expf(-x));
}

// =====================================================================
// Phase 0: zero-pad w_ih (128 x 50) -> (128 x 52) so the GEMM B-operand
// loads need no divergent guards (removes s_and_saveexec tails).
// =====================================================================
__global__ __launch_bounds__(256)
void pad_wih_kernel(const float* __restrict__ w_ih_f,
                    const float* __restrict__ w_ih_r,
                    float* __restrict__ wpad) {      // [2][128][52]
  const int idx = blockIdx.x * 256 + threadIdx.x;
  if (idx >= 2 * G4H * KPAD) return;
  const int dir = idx / (G4H * KPAD);
  const int rem = idx % (G4H * KPAD);
  const int n = rem / KPAD, k = rem % KPAD;
  const float* __restrict__ w = dir ? w_ih_r : w_ih_f;
  wpad[idx] = (k < I_) ? w[n * I_ + k] : 0.0f;
}

// =====================================================================
// Phase A: p[dir][t*B + b][n] = x[b,t,:] @ w_ih[dir]^T + (b_ih+b_hh)
// grid = (T*B/16), block = 512 (16 waves: waves 0-7 dir F, 8-15 dir R)
// =====================================================================
__global__ __launch_bounds__(512)
void lstm_proj_kernel(const float* __restrict__ x,
                      const float* __restrict__ wpad,   // [2][128][52]
                      const float* __restrict__ b_ih_f,
                      const float* __restrict__ b_hh_f,
                      const float* __restrict__ b_ih_r,
                      const float* __restrict__ b_hh_r,
                      float* __restrict__ p) {
  __shared__ float xs[16 * KPAD];           // 16 rows of x, zero-padded K

  const int tid  = threadIdx.x;
  const int m0   = blockIdx.x * 16;         // flat row = t*B + b
  const int t    = m0 >> 9;                 // /512
  const int b0   = m0 & (B_ - 1);

  for (int idx = tid; idx < 16 * KPAD; idx += 512) {
    const int r = idx / KPAD, c = idx % KPAD;
    xs[idx] = (c < I_) ? x[(size_t)(b0 + r) * T_ * I_ + (size_t)t * I_ + c] : 0.0f;
  }
  __syncthreads();

  const int lane = tid & 31;
  const int wave = tid >> 5;                // 0..15
  const int dir  = wave >> 3;               // 0=f, 1=r
  const int n0   = (wave & 7) * 16;
  const int nl   = n0 + (lane & 15);        // output column (0..127)
  const int hh   = lane >> 4;               // lane half

  const float* __restrict__ wp = wpad + (size_t)dir * G4H * KPAD;

  v8f acc = {0.f, 0.f, 0.f, 0.f, 0.f, 0.f, 0.f, 0.f};
  #pragma unroll
  for (int kk = 0; kk < KPAD / 4; ++kk) {   // 13 WMMA steps
    const int k = kk * 4 + 2 * hh;
    v2f a; a.x = xs[(lane & 15) * KPAD + k];
           a.y = xs[(lane & 15) * KPAD + k + 1];
    v2f b; b.x = wp[nl * KPAD + k];
           b.y = wp[nl * KPAD + k + 1];
    acc = wmma_f32_k4(a, b, acc);
  }

  const float bias = dir ? (b_ih_r[nl] + b_hh_r[nl])
                         : (b_ih_f[nl] + b_hh_f[nl]);

  float* __restrict__ pd =
      p + (size_t)dir * T_ * B_ * G4H + (size_t)m0 * G4H;
  #pragma unroll
  for (int r = 0; r < 8; ++r) {
    const int m = r + 8 * hh;
    pd[(size_t)m * G4H + nl] = acc[r] + bias;
  }
}

// =====================================================================
// Phase B: persistent bidirectional LSTM scan with async double-buffered
// projection tiles (GLOBAL_LOAD_ASYNC_TO_LDS_B128, ASYNCcnt-tracked DMA):
// the 8KB tile for step s+1 streams into LDS while step s computes, so
// global latency is off the sequential critical path.
// grid = (B/16, 2), block = 256 (8 waves, wave w owns gate cols 16w..16w+15)
// =====================================================================
#define HS 34    // padded LDS stride for h tile (bank-conflict free)
#define ZS 130   // padded LDS stride for z tile

// per-lane async copy of one 1KB wave-slice of an 8KB tile (2 x b128/lane)
__device__ __forceinline__ void async_tile_load(const float* __restrict__ gsrc,
                                                uint32_t lds_base,
                                                int wave, int lane) {
  const uint32_t loff = lds_base + (uint32_t)(wave * 1024 + lane * 16);
  const char* g = (const char*)gsrc + wave * 1024 + lane * 16;
  asm volatile(
      "global_load_async_to_lds_b128 %0, %1, off\n\t"
      "global_load_async_to_lds_b128 %0, %1, off offset:512"
      :: "v"(loff), "v"(g) : "memory");
}

__global__ __launch_bounds__(256)
void lstm_scan_kernel(const float* __restrict__ p,
                      const float* __restrict__ w_hh_f,
                      const float* __restrict__ w_hh_r,
                      float* __restrict__ hb) {
  __shared__ float z0[2][16 * G4H]; // double-buffered async x-proj tiles
  __shared__ float hbuf[16 * HS];   // current h (16 batch x 32 hidden)
  __shared__ float zbuf[16 * ZS];   // gate pre-activations (16 x 128)

  const int tid  = threadIdx.x;
  const int lane = tid & 31;
  const int wave = tid >> 5;                 // 0..7
  const int dir  = blockIdx.y;
  const int b0   = blockIdx.x * 16;
  const int n0   = wave * 16;
  const int nl   = n0 + (lane & 15);
  const int hh   = lane >> 4;

  const float* __restrict__ whh   = dir ? w_hh_r : w_hh_f;
  const float* __restrict__ pbase = p + (size_t)dir * T_ * B_ * G4H;

  // preload w_hh B-operands (K=32 -> 8 k-steps, 2 VGPRs each) once
  v2f bw[8];
  #pragma unroll
  for (int kk = 0; kk < 8; ++kk) {
    const int k = kk * 4 + 2 * hh;
    bw[kk].x = whh[nl * H_ + k];
    bw[kk].y = whh[nl * H_ + k + 1];
  }

  for (int i = tid; i < 16 * HS; i += 256) hbuf[i] = 0.0f;

  // each lane owns cell state for 2 (batch,hidden) elements
  const int br = tid >> 4;          // batch row in tile (0..15)
  const int j  = (tid & 15) * 2;    // hidden pair (0..30)
  float c0 = 0.0f, c1 = 0.0f;

  const uint32_t zb0 = (uint32_t)(uintptr_t)&z0[0][0];
  const uint32_t zb1 = (uint32_t)(uintptr_t)&z0[1][0];

  // prologue: start DMA of tile for step 0
  {
    const int t0 = dir ? (T_ - 1) : 0;
    async_tile_load(pbase + ((size_t)t0 * B_ + b0) * G4H, zb0, wave, lane);
  }

  for (int s = 0; s < T_; ++s) {
    const int t = dir ? (T_ - 1 - s) : s;
    const int cur = s & 1;

    // kick off DMA for step s+1 into the other buffer, then wait for tile s
    if (s + 1 < T_) {
      const int tn = dir ? (t - 1) : (t + 1);
      async_tile_load(pbase + ((size_t)tn * B_ + b0) * G4H,
                      cur ? zb0 : zb1, wave, lane);
      asm volatile("s_wait_asynccnt 0x2" ::: "memory"); // tile s landed (in-order)
    } else {
      asm volatile("s_wait_asynccnt 0x0" ::: "memory");
    }
    __syncthreads();   // all waves' tile-s slices + previous h visible

    // C init = x-projection tile (already includes both biases)
    const float* __restrict__ zt = &z0[cur][0];
    v8f acc;
    #pragma unroll
    for (int r = 0; r < 8; ++r) acc[r] = zt[(r + 8 * hh) * G4H + nl];

    // z += h @ w_hh^T : 8 fp32 WMMAs over K=32
    #pragma unroll
    for (int kk = 0; kk < 8; ++kk) {
      const int k = kk * 4 + 2 * hh;
      v2f a; a.x = hbuf[(lane & 15) * HS + k];
             a.y = hbuf[(lane & 15) * HS + k + 1];
      acc = wmma_f32_k4(a, bw[kk], acc);
    }

    #pragma unroll
    for (int r = 0; r < 8; ++r) zbuf[(r + 8 * hh) * ZS + nl] = acc[r];
    __syncthreads();   // z complete; all hbuf reads for this step done

    // gate math: i,f,g,o = z[:, 0:32 / 32:64 / 64:96 / 96:128]
    const float* __restrict__ zr = zbuf + br * ZS;
    const float zi0 = zr[j],      zi1 = zr[j + 1];
    const float zf0 = zr[32 + j], zf1 = zr[33 + j];
    const float zg0 = zr[64 + j], zg1 = zr[65 + j];
    const float zo0 = zr[96 + j], zo1 = zr[97 + j];

    c0 = sigm(zf0) * c0 + sigm(zi0) * tanhf(zg0);
    c1 = sigm(zf1) * c1 + sigm(zi1) * tanhf(zg1);
    const float h0 = sigm(zo0) * tanhf(c0);
    const float h1 = sigm(zo1) * tanhf(c1);

    hbuf[br * HS + j]     = h0;
    hbuf[br * HS + j + 1] = h1;

    float* __restrict__ ho =
        hb + ((size_t)(b0 + br) * T_ + t) * (2 * H_) + dir * H_ + j;
    ho[0] = h0; ho[1] = h1;
    // next iteration's top barrier orders: gate reads before zbuf rewrite,
    // hbuf writes before next A-operand reads; z0[cur] reads precede its DMA reuse via the mid-step z barrier
  }
}

// =====================================================================
// Phase C: fused FC head: out = (hb @ fc1_w^T + fc1_b) @ fc3_w^T + fc3_b
// grid = (B*T/16/8), block = 256; each wave owns one 16-row tile
// =====================================================================
#define FS 68   // padded LDS stride for the 16x64 A tile

__global__ __launch_bounds__(256)
void fc_head_kernel(const float* __restrict__ hb,     // (B*T, 64)
                    const float* __restrict__ fc1_w,  // (64, 64)
                    const float* __restrict__ fc1_b,  // (64)
                    const float* __restrict__ fc3_w,  // (64)
                    const float* __restrict__ fc3_b,  // (1)
                    float* __restrict__ out) {        // (B*T)
  __shared__ float tile[8][16 * FS];

  const int tid  = threadIdx.x;
  const int lane = tid & 31;
  const int wave = tid >> 5;                    // 0..7
  const int m0   = (blockIdx.x * 8 + wave) * 16;
  const int hh   = lane >> 4;

  // stage 16x64 A tile: contiguous 4KB, lane-striped float4 (coalesced)
  {
    const float4* __restrict__ src =
        (const float4*)(hb + (size_t)m0 * 64);
    #pragma unroll
    for (int i = 0; i < 8; ++i) {
      const int f = i * 32 + lane;              // float4 index in tile
      const float4 v = src[f];
      const int r = f >> 4, col = (f & 15) * 4;
      *(float4*)&tile[wave][r * FS + col] = v;
    }
  }
  __syncwarp();

  float rowacc[8];
  #pragma unroll
  for (int r = 0; r < 8; ++r) rowacc[r] = 0.0f;

  #pragma unroll
  for (int nt = 0; nt < 4; ++nt) {              // fc1 N tiles
    const int nl = nt * 16 + (lane & 15);
    v8f acc = {0.f, 0.f, 0.f, 0.f, 0.f, 0.f, 0.f, 0.f};
    #pragma unroll
    for (int kk = 0; kk < 16; ++kk) {           // K = 64
      const int k = kk * 4 + 2 * hh;
      v2f a; a.x = tile[wave][(lane & 15) * FS + k];
             a.y = tile[wave][(lane & 15) * FS + k + 1];
      v2f b; b.x = fc1_w[nl * 64 + k];
             b.y = fc1_w[nl * 64 + k + 1];
      acc = wmma_f32_k4(a, b, acc);
    }
    // fuse fc3: scale by fc3 weight, add fc1 bias contribution, reduce over N
    const float w3 = fc3_w[nl];
    const float bc = fc1_b[nl] * w3;
    #pragma unroll
    for (int r = 0; r < 8; ++r) {
      float v = acc[r] * w3 + bc;
      v += __shfl_xor(v, 1, 32);
      v += __shfl_xor(v, 2, 32);
      v += __shfl_xor(v, 4, 32);
      v += __shfl_xor(v, 8, 32);                // sum over 16-lane half
      rowacc[r] += v;
    }
  }

  const float b3 = fc3_b[0];
  if ((lane & 15) == 0) {                       // lane 0 -> rows 0..7, lane 16 -> 8..15
    #pragma unroll
    for (int r = 0; r < 8; ++r)
      out[m0 + 8 * hh + r] = rowacc[r] + b3;
  }
}

// =====================================================================
extern "C" void kernel_launch(void* const* d_in, const int* in_sizes, int n_in,
                              void* d_out, int out_size, void* d_ws, size_t ws_size,
                              hipStream_t stream) {
  const float* x      = (const float*)d_in[0];
  const float* w_ih_f = (const float*)d_in[1];
  const float* w_hh_f = (const float*)d_in[2];
  const float* b_ih_f = (const float*)d_in[3];
  const float* b_hh_f = (const float*)d_in[4];
  const float* w_ih_r = (const float*)d_in[5];
  const float* w_hh_r = (const float*)d_in[6];
  const float* b_ih_r = (const float*)d_in[7];
  const float* b_hh_r = (const float*)d_in[8];
  const float* fc1_w  = (const float*)d_in[9];
  const float* fc1_b  = (const float*)d_in[10];
  const float* fc3_w  = (const float*)d_in[11];
  const float* fc3_b  = (const float*)d_in[12];
  float* out = (float*)d_out;

  // workspace: p (512MB) | hb (128MB) | wpad (52KB)
  float* p    = (float*)d_ws;
  float* hb   = p + (size_t)2 * T_ * B_ * G4H;
  float* wpad = hb + (size_t)B_ * T_ * 2 * H_;

  // Phase 0: pad input-projection weights to K=52
  pad_wih_kernel<<<(2 * G4H * KPAD + 255) / 256, 256, 0, stream>>>(
      w_ih_f, w_ih_r, wpad);

  // Phase A: input projections, both directions
  lstm_proj_kernel<<<(T_ * B_) / 16, 512, 0, stream>>>(
      x, wpad, b_ih_f, b_hh_f, b_ih_r, b_hh_r, p);

  // Phase B: bidirectional recurrent scan (persistent over T, async DMA)
  lstm_scan_kernel<<<dim3(B_ / 16, 2), 256, 0, stream>>>(
      p, w_hh_f, w_hh_r, hb);

  // Phase C: fused FC head
  fc_head_kernel<<<(B_ * T_) / (16 * 8), 256, 0, stream>>>(
      hb, fc1_w, fc1_b, fc3_w, fc3_b, out);
}